// RFCOSPostProcessor_65094524338405
// MI455X (gfx1250) — compile-verified
//
#include <hip/hip_runtime.h>
#include <hip/hip_bf16.h>
#include <stdint.h>

// ---------------------------------------------------------------------------
// RFCOS post-processor for MI455X (gfx1250, wave32).
//
// Per (image, level):
//   score_j = sigmoid(max_c cls[c,j]); gate with sigmoid(ctr_j); thresh 0.2
//   top-1000 by masked score  ->  det[7] = f(loc, reg^3 * nf)
//
// CDNA5 features used (verified in round-1 disasm):
//   * global_load_async_to_lds_b128 + s_wait_asynccnt (ASYNCcnt path) to
//     stage the centerness plane into LDS, 16B per lane per issue.
//   * 320KB-per-WGP LDS: whole level-0 score array (16384 pairs, 128KB) plus
//     the ctr plane (64KB) resident in LDS -> top-K as one in-LDS bitonic
//     sort, zero global round-trips for selection. Impossible on CDNA4's
//     64KB LDS.
//   * b128 vector loads/stores (global_load_b128 / ds_store_b128) on the
//     dominant streaming phase; workload is pure-bandwidth (~65MB @ 23.3TB/s
//     ~ 3us), no matmul content -> WMMA intentionally not used.
// ---------------------------------------------------------------------------

#define N_IMG   32
#define N_CLS   15
#define KOUT    1000
#define NPOS    (5 * KOUT)          // 5000 slots per image
#define THRESH  0.2f

__device__ __forceinline__ float sigm(float x) {
    return 1.0f / (1.0f + __expf(-x));
}

template <int HW>
__launch_bounds__(512)
__global__ void rfcos_level_kernel(const float* __restrict__ loc,   // [HW,2]
                                   const float* __restrict__ cls,   // [32,15,HW]
                                   const float* __restrict__ reg,   // [32,5,HW]
                                   const float* __restrict__ ctr,   // [32,1,HW]
                                   float* __restrict__ out,
                                   float nf, int level)
{
    // Dynamic LDS layout (bytes):
    //   [0,        HW*4)  : s_val  (masked scores)
    //   [HW*4,     HW*8)  : s_idx  (position | label<<20)
    //   [HW*8,     HW*12) : s_ctr  (async-staged centerness logits)
    extern __shared__ char smem[];
    float* s_val = reinterpret_cast<float*>(smem);
    int*   s_idx = reinterpret_cast<int*>(smem + (size_t)HW * 4);
    float* s_ctr = reinterpret_cast<float*>(smem + (size_t)HW * 8);

    const int n   = blockIdx.x;      // image
    const int T   = blockDim.x;      // min(512, HW)
    const int tid = threadIdx.x;

    // ---- Phase 1: async-copy ctr plane global -> LDS, 16B/lane (ASYNCcnt) -
    const float* ctrN = ctr + (size_t)n * HW;
    for (int j0 = tid * 4; j0 < HW; j0 += T * 4) {
        unsigned lds_off = (unsigned)((size_t)HW * 8 + (size_t)j0 * 4);
        unsigned long long ga = (unsigned long long)(uintptr_t)(ctrN + j0);
        asm volatile("global_load_async_to_lds_b128 %0, %1, off"
                     :: "v"(lds_off), "v"(ga) : "memory");
    }
#if defined(__has_builtin) && __has_builtin(__builtin_amdgcn_s_wait_asynccnt)
    __builtin_amdgcn_s_wait_asynccnt(0);
#else
    asm volatile("s_wait_asynccnt 0" ::: "memory");
#endif
    __syncthreads();

    // ---- Phase 2: score / argmax / gate, 4 positions per thread, b128 -----
    // sigmoid is monotonic: max(sigmoid(x_c)) == sigmoid(max(x_c)), same argmax.
    const float* clsN = cls + (size_t)n * N_CLS * HW;
    for (int j0 = tid * 4; j0 < HW; j0 += T * 4) {
        float4 best = *reinterpret_cast<const float4*>(clsN + j0);
        int bx = 0, by = 0, bz = 0, bw = 0;
#pragma unroll
        for (int c = 1; c < N_CLS; ++c) {
            float4 v = *reinterpret_cast<const float4*>(clsN + (size_t)c * HW + j0);
            if (v.x > best.x) { best.x = v.x; bx = c; }
            if (v.y > best.y) { best.y = v.y; by = c; }
            if (v.z > best.z) { best.z = v.z; bz = c; }
            if (v.w > best.w) { best.w = v.w; bw = c; }
        }
        float4 cl = *reinterpret_cast<const float4*>(s_ctr + j0);   // ds_load_b128

        float4 p;
        p.x = sigm(best.x); p.y = sigm(best.y);
        p.z = sigm(best.z); p.w = sigm(best.w);

        float4 val;
        val.x = (p.x * sigm(cl.x)) > THRESH ? p.x : -1.0f;
        val.y = (p.y * sigm(cl.y)) > THRESH ? p.y : -1.0f;
        val.z = (p.z * sigm(cl.z)) > THRESH ? p.z : -1.0f;
        val.w = (p.w * sigm(cl.w)) > THRESH ? p.w : -1.0f;

        int4 idx;
        idx.x = (j0 + 0) | (bx << 20);           // pack label (j < 2^20)
        idx.y = (j0 + 1) | (by << 20);
        idx.z = (j0 + 2) | (bz << 20);
        idx.w = (j0 + 3) | (bw << 20);

        *reinterpret_cast<float4*>(s_val + j0) = val;   // ds_store_b128
        *reinterpret_cast<int4*>(s_idx + j0)   = idx;   // ds_store_b128
    }
    __syncthreads();

    // ---- Phase 3: in-LDS bitonic sort, descending, index tie-break --------
    for (int k = 2; k <= HW; k <<= 1) {
        for (int jj = k >> 1; jj > 0; jj >>= 1) {
            for (int t = tid; t < HW / 2; t += T) {
                int i  = ((t & ~(jj - 1)) << 1) | (t & (jj - 1));
                int p2 = i | jj;
                bool descend = ((i & k) == 0);
                float a = s_val[i], b = s_val[p2];
                int  ia = s_idx[i], ib = s_idx[p2];
                bool a_first = (a > b) ||
                               (a == b && (ia & 0xFFFFF) < (ib & 0xFFFFF));
                if (a_first != descend) {
                    s_val[i] = b; s_val[p2] = a;
                    s_idx[i] = ib; s_idx[p2] = ia;
                }
            }
            __syncthreads();
        }
    }

    // ---- Phase 4: emit top-K (padded, masked) -----------------------------
    constexpr int KEFF = (HW < KOUT) ? HW : KOUT;
    float* detO = out;                                  // [32][5000][7]
    float* scO  = out + (size_t)N_IMG * NPOS * 7;       // [32][5000]
    float* lbO  = scO + (size_t)N_IMG * NPOS;           // [32][5000]
    float* vaO  = lbO + (size_t)N_IMG * NPOS;           // [32][5000]

    const float* regN = reg + (size_t)n * 5 * HW;
    for (int s = tid; s < KOUT; s += T) {
        float d0=0,d1=0,d2=0,d3=0,d4=0,d5=0,d6=0, sc=0, lb=0, va=0;
        if (s < KEFF) {
            float v   = s_val[s];
            int  code = s_idx[s];
            if (v > 0.0f) {                      // == candidate mask
                int j = code & 0xFFFFF;
                int c = code >> 20;
                float r[5];
#pragma unroll
                for (int q = 0; q < 5; ++q) {
                    float t2 = regN[(size_t)q * HW + j];
                    r[q] = t2 * t2 * t2 * nf;    // pow(.,3) * normal_factor
                }
                float x = loc[2 * j], y = loc[2 * j + 1];
                d0 = x;        d1 = y;
                d2 = x - r[0]; d3 = y - r[1];
                d4 = x - r[2]; d5 = y - r[3];
                d6 = r[4];
                sc = v; lb = (float)(c + 1); va = 1.0f;
            }
        }
        size_t pos = (size_t)n * NPOS + (size_t)level * KOUT + s;
        float* dp = detO + pos * 7;
        dp[0]=d0; dp[1]=d1; dp[2]=d2; dp[3]=d3; dp[4]=d4; dp[5]=d5; dp[6]=d6;
        scO[pos] = sc;
        lbO[pos] = lb;
        vaO[pos] = va;
    }
}

extern "C" void kernel_launch(void* const* d_in, const int* in_sizes, int n_in,
                              void* d_out, int out_size, void* d_ws, size_t ws_size,
                              hipStream_t stream) {
    (void)in_sizes; (void)n_in; (void)out_size; (void)d_ws; (void)ws_size;
    static const float NF[5] = {16.f, 32.f, 64.f, 128.f, 256.f};
    float* out = (float*)d_out;

    for (int l = 0; l < 5; ++l) {
        const float* loc = (const float*)d_in[4 * l + 0];
        const float* cls = (const float*)d_in[4 * l + 1];
        const float* reg = (const float*)d_in[4 * l + 2];
        const float* ctr = (const float*)d_in[4 * l + 3];
        switch (l) {
        case 0: rfcos_level_kernel<16384><<<N_IMG, 512, 16384u * 12u, stream>>>(
                    loc, cls, reg, ctr, out, NF[l], l); break;
        case 1: rfcos_level_kernel<4096><<<N_IMG, 512, 4096u * 12u, stream>>>(
                    loc, cls, reg, ctr, out, NF[l], l); break;
        case 2: rfcos_level_kernel<1024><<<N_IMG, 512, 1024u * 12u, stream>>>(
                    loc, cls, reg, ctr, out, NF[l], l); break;
        case 3: rfcos_level_kernel<256><<<N_IMG, 256, 256u * 12u, stream>>>(
                    loc, cls, reg, ctr, out, NF[l], l); break;
        case 4: rfcos_level_kernel<64><<<N_IMG, 64, 64u * 12u, stream>>>(
                    loc, cls, reg, ctr, out, NF[l], l); break;
        }
    }
}